// GATTopNet_39109972198055
// MI455X (gfx1250) — compile-verified
//
#include <hip/hip_runtime.h>

// GATTopNet forward for MI455X (gfx1250, wave32).
// - Node GEMMs + edge readout MLP (~139 GFLOP fp32) on V_WMMA_F32_16X16X4_F32,
//   with 4-way M-tile blocking so each B-fragment load feeds 4 WMMA issues.
// - Edge softmax/aggregation done as atomic-free gathers over a dst-CSR built
//   once per call; per-node edge lists sorted -> fully deterministic fp sums.

#define NN 50000      // nodes
#define NE 1600000    // edges (divisible by 64)

typedef __attribute__((ext_vector_type(2))) float v2f;
typedef __attribute__((ext_vector_type(8))) float v8f;

__device__ __forceinline__ v8f wmma4(v2f a, v2f b, v8f c) {
    return __builtin_amdgcn_wmma_f32_16x16x4_f32(false, a, false, b,
                                                 (short)0, c, false, false);
}

// ---------------------------------------------------------------------------
// C[M x 128] = A[M x 128] @ B[128 x 128] (+ bias). 64 rows per block
// (4 M-tiles per wave -> each B fragment reused 4x), 8 waves own 16-col
// stripes. A tile staged in LDS; tail block guarded.
// ---------------------------------------------------------------------------
__global__ __launch_bounds__(256) void gemm_node(
    const float* __restrict__ A, const float* __restrict__ B,
    const float* __restrict__ bias, float* __restrict__ C, int M) {
    __shared__ __align__(16) float As[64 * 128];   // 32 KB
    const int t = threadIdx.x;
    const int rowBase = blockIdx.x * 64;
    #pragma unroll
    for (int i = 0; i < 8; ++i) {                  // 8192 floats, 8 x float4/thread
        int e = (t + i * 256) * 4;
        int r = e >> 7, c = e & 127;
        int row = rowBase + r;
        float4 v;
        if (row < M) v = *(const float4*)&A[(size_t)row * 128 + c];
        else { v.x = v.y = v.z = v.w = 0.f; }
        *(float4*)&As[e] = v;
    }
    __syncthreads();

    const int wave = t >> 5, lane = t & 31, half = lane >> 4, lr = lane & 15;
    const int col = wave * 16 + lr;
    v8f acc[4] = {{}, {}, {}, {}};
    #pragma unroll 4
    for (int k0 = 0; k0 < 128; k0 += 4) {
        int ka = k0 + 2 * half;
        v2f b = { B[(size_t)ka * 128 + col], B[(size_t)(ka + 1) * 128 + col] };
        #pragma unroll
        for (int mt = 0; mt < 4; ++mt) {
            const float* ap = &As[(mt * 16 + lr) * 128 + ka];
            v2f a = { ap[0], ap[1] };
            acc[mt] = wmma4(a, b, acc[mt]);
        }
    }
    const float bc = bias ? bias[col] : 0.0f;
    #pragma unroll
    for (int mt = 0; mt < 4; ++mt)
        #pragma unroll
        for (int v = 0; v < 8; ++v) {
            int row = rowBase + mt * 16 + v + 8 * half;
            if (row < M) C[(size_t)row * 128 + col] = acc[mt][v] + bc;
        }
}

// ---------------------------------------------------------------------------
// CSR-by-dst construction (once per call; src/dst shared by all layers)
// ---------------------------------------------------------------------------
__global__ void zero_i32(int* __restrict__ p, int n) {
    int t = blockIdx.x * blockDim.x + threadIdx.x;
    if (t < n) p[t] = 0;
}

__global__ void edge_deg(const int* __restrict__ dst, int* __restrict__ deg) {
    int e = blockIdx.x * blockDim.x + threadIdx.x;
    if (e < NE) atomicAdd(&deg[dst[e]], 1);
}

__global__ __launch_bounds__(1024) void scan_rowptr(const int* __restrict__ deg,
                                                    int* __restrict__ rowp) {
    __shared__ int part[1024];
    const int t = threadIdx.x;
    const int CH = (NN + 1023) / 1024;
    const int base = t * CH;
    int sum = 0;
    for (int i = 0; i < CH; ++i) {
        int idx = base + i;
        if (idx < NN) sum += deg[idx];
    }
    part[t] = sum;
    __syncthreads();
    for (int off = 1; off < 1024; off <<= 1) {
        int v = part[t];
        int add = (t >= off) ? part[t - off] : 0;
        __syncthreads();
        part[t] = v + add;
        __syncthreads();
    }
    int run = (t == 0) ? 0 : part[t - 1];
    for (int i = 0; i < CH; ++i) {
        int idx = base + i;
        if (idx < NN) { rowp[idx] = run; run += deg[idx]; }
    }
    if (t == 1023) rowp[NN] = part[1023];
}

__global__ void edge_fill(const int* __restrict__ dst,
                          const int* __restrict__ rowp,
                          int* __restrict__ cursor, int* __restrict__ eidx) {
    int e = blockIdx.x * blockDim.x + threadIdx.x;
    if (e >= NE) return;
    int d = dst[e];
    int pos = atomicAdd(&cursor[d], 1);
    eidx[rowp[d] + pos] = e;
}

__global__ void sort_lists(const int* __restrict__ rowp, int* __restrict__ eidx) {
    int n = blockIdx.x * blockDim.x + threadIdx.x;
    if (n >= NN) return;
    int b = rowp[n], e = rowp[n + 1];
    for (int i = b + 1; i < e; ++i) {
        int key = eidx[i];
        int j = i - 1;
        while (j >= b && eidx[j] > key) { eidx[j + 1] = eidx[j]; --j; }
        eidx[j + 1] = key;
    }
}

// ---------------------------------------------------------------------------
// GAT layer pieces (gather-only, deterministic, no fp atomics)
// ---------------------------------------------------------------------------
__global__ void node_attn(const float* __restrict__ ft,
                          const float* __restrict__ al,
                          const float* __restrict__ ar,
                          float* __restrict__ el, float* __restrict__ er,
                          int H, int D) {
    int t = blockIdx.x * blockDim.x + threadIdx.x;
    if (t >= NN * H) return;
    int n = t / H, h = t % H;
    const float* f = ft + (size_t)n * 128 + h * D;
    const float* a = al + h * D;
    const float* b = ar + h * D;
    float sl = 0.f, sr = 0.f;
    for (int d = 0; d < D; ++d) { sl += f[d] * a[d]; sr += f[d] * b[d]; }
    el[t] = sl; er[t] = sr;
}

__global__ void node_msoft(const float* __restrict__ el,
                           const float* __restrict__ er,
                           const int* __restrict__ src,
                           const int* __restrict__ rowp,
                           const int* __restrict__ eidx,
                           float* __restrict__ m, float* __restrict__ s, int H) {
    int t = blockIdx.x * blockDim.x + threadIdx.x;
    if (t >= NN * H) return;
    int n = t / H, h = t % H;
    int b = rowp[n], e = rowp[n + 1];
    float ern = er[t];
    float mx = -1e30f;
    for (int j = b; j < e; ++j) {
        float v = el[src[eidx[j]] * H + h] + ern;
        v = v > 0.f ? v : 0.2f * v;
        mx = fmaxf(mx, v);
    }
    float ss = 0.f;
    for (int j = b; j < e; ++j) {
        float v = el[src[eidx[j]] * H + h] + ern;
        v = v > 0.f ? v : 0.2f * v;
        ss += __expf(v - mx);
    }
    m[t] = mx; s[t] = ss;
}

__global__ void edge_coef(const float* __restrict__ el,
                          const float* __restrict__ er,
                          const int* __restrict__ src,
                          const int* __restrict__ dst,
                          const float* __restrict__ m,
                          const float* __restrict__ s,
                          float* __restrict__ ebuf, int H) {
    int t = blockIdx.x * blockDim.x + threadIdx.x;
    if (t >= NE * H) return;
    int e = t / H, h = t % H;
    int d = dst[e];
    float v = el[src[e] * H + h] + er[d * H + h];
    v = v > 0.f ? v : 0.2f * v;
    ebuf[t] = __expf(v - m[d * H + h]) / s[d * H + h];
}

__global__ void node_aggr(const float* __restrict__ ebuf,
                          const float* __restrict__ ft,
                          const int* __restrict__ src,
                          const int* __restrict__ rowp,
                          const int* __restrict__ eidx,
                          float* __restrict__ x, int H, int D) {
    int t = blockIdx.x * blockDim.x + threadIdx.x;
    if (t >= NN * 128) return;
    int n = t >> 7, dim = t & 127;
    int h = dim / D;
    int b = rowp[n], e = rowp[n + 1];
    float acc = 0.f;
    for (int j = b; j < e; ++j) {
        int ed = eidx[j];
        acc += ebuf[(size_t)ed * H + h] * ft[(size_t)src[ed] * 128 + dim];
    }
    x[t] += (acc > 0.f) ? acc : (__expf(acc) - 1.0f);
}

// ---------------------------------------------------------------------------
// Edge readout MLP, fused per 64-edge tile (WMMA f32, 4-way B reuse).
// LDS phase overlay: [he 64KB | y1 32KB], y2 (16KB) overlays dead he.
// ---------------------------------------------------------------------------
__global__ __launch_bounds__(256) void readout(
    const float* __restrict__ x, const int* __restrict__ src,
    const int* __restrict__ dst,
    const float* __restrict__ W1, const float* __restrict__ b1,
    const float* __restrict__ W2, const float* __restrict__ b2,
    const float* __restrict__ W3, const float* __restrict__ b3,
    float* __restrict__ out) {
    __shared__ __align__(16) float smem[64 * 256 + 64 * 128];   // 96 KB
    float* he = smem;              // [64][256]
    float* y1 = smem + 64 * 256;   // [64][128]
    float* y2 = smem;              // [64][64], overlays he after layer 1
    const int t = threadIdx.x;
    const int e0 = blockIdx.x * 64;

    {   // stage he: thread t loads 64 contiguous floats of one edge row
        int row = t >> 2;
        int c0 = (t & 3) * 64;     // chunk stays within one (src|dst) half
        int ei = e0 + row;
        const float* p = (c0 < 128)
            ? &x[(size_t)src[ei] * 128 + c0]
            : &x[(size_t)dst[ei] * 128 + (c0 - 128)];
        float* d = &he[row * 256 + c0];
        #pragma unroll
        for (int i = 0; i < 64; i += 4) *(float4*)&d[i] = *(const float4*)&p[i];
    }
    __syncthreads();

    const int wave = t >> 5, lane = t & 31, half = lane >> 4, lr = lane & 15;

    {   // layer 1: y1 = relu(he @ W1 + b1); N=128, K=256, 4 M-tiles/wave
        int col = wave * 16 + lr;
        v8f acc[4] = {{}, {}, {}, {}};
        #pragma unroll 4
        for (int k0 = 0; k0 < 256; k0 += 4) {
            int ka = k0 + 2 * half;
            v2f b = { W1[(size_t)ka * 128 + col], W1[(size_t)(ka + 1) * 128 + col] };
            #pragma unroll
            for (int mt = 0; mt < 4; ++mt) {
                const float* ap = &he[(mt * 16 + lr) * 256 + ka];
                v2f a = { ap[0], ap[1] };
                acc[mt] = wmma4(a, b, acc[mt]);
            }
        }
        float bb = b1[col];
        #pragma unroll
        for (int mt = 0; mt < 4; ++mt)
            #pragma unroll
            for (int v = 0; v < 8; ++v) {
                float val = acc[mt][v] + bb;
                y1[(mt * 16 + v + 8 * half) * 128 + col] = val > 0.f ? val : 0.f;
            }
    }
    __syncthreads();

    {   // layer 2: y2 = relu(y1 @ W2 + b2); N=64, K=128
        // waves 0-3: M-tiles 0-1; waves 4-7: M-tiles 2-3 (all waves busy)
        int cw = wave & 3;
        int mtBase = (wave >> 2) * 2;
        int col = cw * 16 + lr;
        v8f acc[2] = {{}, {}};
        #pragma unroll 4
        for (int k0 = 0; k0 < 128; k0 += 4) {
            int ka = k0 + 2 * half;
            v2f b = { W2[(size_t)ka * 64 + col], W2[(size_t)(ka + 1) * 64 + col] };
            #pragma unroll
            for (int mt = 0; mt < 2; ++mt) {
                const float* ap = &y1[((mtBase + mt) * 16 + lr) * 128 + ka];
                v2f a = { ap[0], ap[1] };
                acc[mt] = wmma4(a, b, acc[mt]);
            }
        }
        float bb = b2[col];
        #pragma unroll
        for (int mt = 0; mt < 2; ++mt)
            #pragma unroll
            for (int v = 0; v < 8; ++v) {
                float val = acc[mt][v] + bb;
                y2[((mtBase + mt) * 16 + v + 8 * half) * 64 + col] =
                    val > 0.f ? val : 0.f;
            }
    }
    __syncthreads();

    if (t < 128) {   // layer 3: 64 edges x 2 classes
        int er = t >> 1, cls = t & 1;
        float sum = b3[cls];
        #pragma unroll 8
        for (int k = 0; k < 64; ++k) sum += y2[er * 64 + k] * W3[k * 2 + cls];
        out[(size_t)(e0 + er) * 2 + cls] = sum;
    }
}

// ---------------------------------------------------------------------------
extern "C" void kernel_launch(void* const* d_in, const int* in_sizes, int n_in,
                              void* d_out, int out_size, void* d_ws, size_t ws_size,
                              hipStream_t stream) {
    (void)in_sizes; (void)n_in; (void)out_size; (void)ws_size;
    const float* h     = (const float*)d_in[0];
    const int*   src   = (const int*)  d_in[1];
    const int*   dst   = (const int*)  d_in[2];
    const float* W_emb = (const float*)d_in[3];
    const float* b_emb = (const float*)d_in[4];
    const float* W_g   = (const float*)d_in[5];
    const float* al_g  = (const float*)d_in[6];
    const float* ar_g  = (const float*)d_in[7];
    const float* W_l   = (const float*)d_in[8];
    const float* al_l  = (const float*)d_in[9];
    const float* ar_l  = (const float*)d_in[10];
    const float* W1    = (const float*)d_in[11];
    const float* b1    = (const float*)d_in[12];
    const float* W2    = (const float*)d_in[13];
    const float* b2    = (const float*)d_in[14];
    const float* W3    = (const float*)d_in[15];
    const float* b3    = (const float*)d_in[16];
    float* out = (float*)d_out;

    // workspace carve (~116 MB)
    char* w = (char*)d_ws;
    auto carve = [&](size_t bytes) { char* p = w; w += (bytes + 255) & ~size_t(255); return p; };
    float* x    = (float*)carve((size_t)NN * 128 * 4);
    float* ft   = (float*)carve((size_t)NN * 128 * 4);
    float* el   = (float*)carve((size_t)NN * 8 * 4);
    float* er   = (float*)carve((size_t)NN * 8 * 4);
    float* m    = (float*)carve((size_t)NN * 8 * 4);
    float* s    = (float*)carve((size_t)NN * 8 * 4);
    float* ebuf = (float*)carve((size_t)NE * 8 * 4);
    int* deg    = (int*)carve((size_t)NN * 4);
    int* cursor = (int*)carve((size_t)NN * 4);
    int* rowp   = (int*)carve((size_t)(NN + 1) * 4);
    int* eidx   = (int*)carve((size_t)NE * 4);

    const int gemmBlocks = (NN + 63) / 64;       // 782 (tail guarded)
    const int nGrid      = (NN + 255) / 256;
    const int eGrid      = (NE + 255) / 256;
    const int nfGrid     = (NN * 128 + 255) / 256;

    // ---- CSR by dst (once; reused by all 4 layers) ----
    zero_i32<<<nGrid, 256, 0, stream>>>(deg, NN);
    zero_i32<<<nGrid, 256, 0, stream>>>(cursor, NN);
    edge_deg<<<eGrid, 256, 0, stream>>>(dst, deg);
    scan_rowptr<<<1, 1024, 0, stream>>>(deg, rowp);
    edge_fill<<<eGrid, 256, 0, stream>>>(dst, rowp, cursor, eidx);
    sort_lists<<<nGrid, 256, 0, stream>>>(rowp, eidx);

    // ---- embedding: x = h @ W_emb + b_emb ----
    gemm_node<<<gemmBlocks, 256, 0, stream>>>(h, W_emb, b_emb, x, NN);

    auto run_layer = [&](const float* W, const float* al, const float* ar,
                         int H, int D) {
        gemm_node<<<gemmBlocks, 256, 0, stream>>>(x, W, nullptr, ft, NN);
        int nh = NN * H;
        node_attn<<<(nh + 255) / 256, 256, 0, stream>>>(ft, al, ar, el, er, H, D);
        node_msoft<<<(nh + 255) / 256, 256, 0, stream>>>(el, er, src, rowp, eidx,
                                                         m, s, H);
        int eh = NE * H;
        edge_coef<<<(eh + 255) / 256, 256, 0, stream>>>(el, er, src, dst, m, s,
                                                        ebuf, H);
        node_aggr<<<nfGrid, 256, 0, stream>>>(ebuf, ft, src, rowp, eidx, x, H, D);
    };

    for (int i = 0; i < 3; ++i)
        run_layer(W_g + (size_t)i * 128 * 128, al_g + i * 128, ar_g + i * 128,
                  /*H=*/8, /*D=*/16);
    run_layer(W_l, al_l, ar_l, /*H=*/1, /*D=*/128);

    readout<<<NE / 64, 256, 0, stream>>>(x, src, dst, W1, b1, W2, b2, W3, b3, out);
}